// GPO_25366076850502
// MI455X (gfx1250) — compile-verified
//
#include <hip/hip_runtime.h>
#include <hip/hip_bf16.h>
#include <math.h>

// ---------------------------------------------------------------------------
// Sizes fixed by the reference: B=64, K=512, D=1024, H=32, D_PE=32, 3H=96.
// d_out = pooled[64*1024] ++ weights[64*512]  (float32)
// ---------------------------------------------------------------------------

#define BB 64
#define KK 512
#define DD 1024
#define HH 32
#define G3H 96   // 3*H
#define DPE 32

typedef __attribute__((ext_vector_type(2))) float v2f;
typedef __attribute__((ext_vector_type(8))) float v8f;

__device__ __forceinline__ float pe_val(int p, int c) {
    // pe[p][2i]   = sin(p * div_i),  pe[p][2i+1] = cos(p * div_i)
    // div_i = exp(-(2i) * ln(10000)/32)
    int i2 = (c >> 1) * 2;
    float div = __expf((float)i2 * (-9.210340371976184f / 32.0f)); // ln(10000)=9.2103
    float ang = (float)p * div;
    return (c & 1) ? __cosf(ang) : __sinf(ang);
}

// ---------------------------------------------------------------------------
// Kernel 1: xproj[dir][k][0..95] = pe[k][0..31] @ kernel[32x96] + bias0
// One wave (32 threads) per block. 64 blocks = 2 dirs x 32 M-tiles.
// Uses V_WMMA_F32_16X16X4_F32: per 16x16 output tile, 8 chained K=4 steps.
// ---------------------------------------------------------------------------
__global__ __launch_bounds__(32) void gpo_xproj_wmma(
        const float* __restrict__ kernF, const float* __restrict__ biasF,
        const float* __restrict__ kernB, const float* __restrict__ biasB,
        float* __restrict__ xprojF, float* __restrict__ xprojB) {
    const int dir   = blockIdx.x >> 5;     // 0 = fwd, 1 = bwd
    const int mtile = blockIdx.x & 31;     // 32 tiles of 16 rows (K positions)
    const float* kern = dir ? kernB : kernF;
    const float* bias = dir ? biasB : biasF;   // row 0 of the [2,96] bias
    float* xout = dir ? xprojB : xprojF;

    const int lane  = threadIdx.x;
    const int mrow  = lane & 15;
    const int khalf = lane >> 4;           // 0: K pair {0,1}, 1: K pair {2,3}
    const int row   = mtile * 16 + mrow;   // pe row (= sequence position)

    // A fragments (16x4 f32, M=pe rows, K=pe channels), all 8 K-steps.
    v2f afrag[8];
    #pragma unroll
    for (int kk = 0; kk < 8; ++kk) {
        int c0 = kk * 4 + khalf * 2;
        afrag[kk].x = pe_val(row, c0);
        afrag[kk].y = pe_val(row, c0 + 1);
    }

    const int ncol = lane & 15;
    #pragma unroll
    for (int nt = 0; nt < 6; ++nt) {       // 6 tiles of 16 output columns
        v8f acc = {};
        #pragma unroll
        for (int kk = 0; kk < 8; ++kk) {
            int kr = kk * 4 + khalf * 2;
            int col = nt * 16 + ncol;
            v2f bfrag;
            bfrag.x = kern[(kr + 0) * G3H + col];
            bfrag.y = kern[(kr + 1) * G3H + col];
            acc = __builtin_amdgcn_wmma_f32_16x16x4_f32(
                false, afrag[kk], false, bfrag, (short)0, acc, false, false);
        }
        // D layout: VGPR r, lane<16 -> M=r ; lane>=16 -> M=8+r ; N = lane%16
        #pragma unroll
        for (int r = 0; r < 8; ++r) {
            int orow = mtile * 16 + r + khalf * 8;
            int ocol = nt * 16 + ncol;
            xout[orow * G3H + ocol] = acc[r] + bias[ocol];
        }
    }
}

// ---------------------------------------------------------------------------
// Kernel 2: per-batch GRU recurrence (fwd + bwd), score accumulation and
// masked softmax. 64 blocks (one per batch), 96 threads.
// ---------------------------------------------------------------------------
__device__ __forceinline__ float sigmoidf_(float x) {
    return 1.0f / (1.0f + expf(-x));
}

__global__ __launch_bounds__(96) void gpo_gru_softmax(
        const float* __restrict__ xprojF, const float* __restrict__ xprojB,
        const float* __restrict__ rkF, const float* __restrict__ bF,
        const float* __restrict__ rkB, const float* __restrict__ bB,
        const float* __restrict__ lw, const int* __restrict__ lengths,
        float* __restrict__ wout /* d_out + B*D, [B,K] */) {
    __shared__ float rk[HH * G3H];   // 12 KB recurrent kernel
    __shared__ float h[HH];
    __shared__ float hp[G3H];
    __shared__ float score[KK];
    __shared__ float red[G3H];

    const int b = blockIdx.x;
    const int t = threadIdx.x;
    const int L = lengths[b];

    for (int i = t; i < KK; i += G3H) score[i] = 0.0f;

    // ---------------- forward direction ----------------
    for (int i = t; i < HH * G3H; i += G3H) rk[i] = rkF[i];
    if (t < HH) h[t] = 0.0f;
    __syncthreads();
    {
        const float bias1 = bF[G3H + t];
        for (int k = 0; k < KK; ++k) {
            if (k < L) {   // uniform branch per block
                float acc = bias1;
                #pragma unroll 8
                for (int j = 0; j < HH; ++j) acc += h[j] * rk[j * G3H + t];
                hp[t] = acc;
                __syncthreads();
                if (t < HH) {
                    const float* xp = xprojF + k * G3H;
                    float z  = sigmoidf_(xp[t]        + hp[t]);
                    float r  = sigmoidf_(xp[HH + t]   + hp[HH + t]);
                    float hc = tanhf    (xp[2*HH + t] + r * hp[2*HH + t]);
                    h[t] = z * h[t] + (1.0f - z) * hc;
                }
                __syncthreads();
            }
            if (t == 0) {
                float s = 0.0f;
                for (int u = 0; u < HH; ++u) s += h[u] * lw[u];
                score[k] += s;
            }
        }
    }
    __syncthreads();

    // ---------------- backward direction ----------------
    for (int i = t; i < HH * G3H; i += G3H) rk[i] = rkB[i];
    if (t < HH) h[t] = 0.0f;
    __syncthreads();
    {
        const float bias1 = bB[G3H + t];
        for (int st = 0; st < KK; ++st) {
            const int j = (KK - 1) - st;      // original position
            if (j < L) {                       // flipped mask, uniform
                float acc = bias1;
                #pragma unroll 8
                for (int jj = 0; jj < HH; ++jj) acc += h[jj] * rk[jj * G3H + t];
                hp[t] = acc;
                __syncthreads();
                if (t < HH) {
                    const float* xp = xprojB + j * G3H;
                    float z  = sigmoidf_(xp[t]        + hp[t]);
                    float r  = sigmoidf_(xp[HH + t]   + hp[HH + t]);
                    float hc = tanhf    (xp[2*HH + t] + r * hp[2*HH + t]);
                    h[t] = z * h[t] + (1.0f - z) * hc;
                }
                __syncthreads();
            }
            if (t == 0) {
                float s = 0.0f;
                for (int u = 0; u < HH; ++u) s += h[u] * lw[HH + u];
                score[j] += s;
            }
        }
    }
    __syncthreads();

    // ---------------- masked softmax over K (temp 0.1) ----------------
    for (int k = t; k < KK; k += G3H)
        score[k] = ((k < L) ? score[k] : -10000.0f) * 10.0f;
    __syncthreads();

    float lm = -3.0e38f;
    for (int k = t; k < KK; k += G3H) lm = fmaxf(lm, score[k]);
    red[t] = lm;
    __syncthreads();
    if (t == 0) {
        float m = red[0];
        for (int i = 1; i < G3H; ++i) m = fmaxf(m, red[i]);
        red[0] = m;
    }
    __syncthreads();
    const float m = red[0];
    __syncthreads();

    float ls = 0.0f;
    for (int k = t; k < KK; k += G3H) {
        float e = expf(score[k] - m);
        score[k] = e;
        ls += e;
    }
    red[t] = ls;
    __syncthreads();
    if (t == 0) {
        float s = 0.0f;
        for (int i = 0; i < G3H; ++i) s += red[i];
        red[0] = s;
    }
    __syncthreads();
    const float inv = 1.0f / red[0];
    for (int k = t; k < KK; k += G3H)
        wout[b * KK + k] = score[k] * inv;
}

// ---------------------------------------------------------------------------
// Kernel 3: per-(b,d) descending bitonic sort of the masked 512-column and
// weighted sum with softmax weights. 65536 blocks, 256 threads, 3KB LDS.
// ---------------------------------------------------------------------------
__global__ __launch_bounds__(256) void gpo_sort_pool(
        const float* __restrict__ feat, const int* __restrict__ lengths,
        const float* __restrict__ weights /* d_out + B*D */,
        float* __restrict__ pooled /* d_out */) {
    __shared__ float s[KK];
    __shared__ float red[256];

    const int bd = blockIdx.x;
    const int b = bd >> 10;          // / 1024
    const int d = bd & 1023;
    const int L = lengths[b];
    const int tid = threadIdx.x;

    for (int k = tid; k < KK; k += 256) {
        float v = feat[((size_t)b * KK + k) * DD + d];
        s[k] = (k < L) ? v : -3.0e38f;   // sentinel: padded slots sink to the end
    }
    __syncthreads();

    // Bitonic sort, overall descending.
    for (int sz = 2; sz <= KK; sz <<= 1) {
        for (int st = sz >> 1; st > 0; st >>= 1) {
            for (int i = tid; i < KK; i += 256) {
                int ix = i ^ st;
                if (ix > i) {
                    float a = s[i], c = s[ix];
                    bool desc = ((i & sz) == 0);
                    if (desc ? (a < c) : (a > c)) { s[i] = c; s[ix] = a; }
                }
            }
            __syncthreads();
        }
    }

    // pooled[b,d] = sum_{k<L} sorted[k] * weights[b,k]
    float acc = 0.0f;
    for (int k = tid; k < L; k += 256) acc += s[k] * weights[b * KK + k];
    red[tid] = acc;
    __syncthreads();
    for (int off = 128; off > 0; off >>= 1) {
        if (tid < off) red[tid] += red[tid + off];
        __syncthreads();
    }
    if (tid == 0) pooled[(size_t)b * DD + d] = red[0];
}

// ---------------------------------------------------------------------------
// Host launcher
// ---------------------------------------------------------------------------
extern "C" void kernel_launch(void* const* d_in, const int* in_sizes, int n_in,
                              void* d_out, int out_size, void* d_ws, size_t ws_size,
                              hipStream_t stream) {
    const float* features  = (const float*)d_in[0];   // [64,512,1024]
    const int*   lengths   = (const int*)  d_in[1];   // [64]
    const float* fwd_k     = (const float*)d_in[2];   // [32,96]
    const float* fwd_rk    = (const float*)d_in[3];   // [32,96]
    const float* fwd_b     = (const float*)d_in[4];   // [2,96]
    const float* bwd_k     = (const float*)d_in[5];
    const float* bwd_rk    = (const float*)d_in[6];
    const float* bwd_b     = (const float*)d_in[7];
    const float* linear_w  = (const float*)d_in[8];   // [64,1] flat

    float* pooled  = (float*)d_out;                   // [64,1024]
    float* wout    = (float*)d_out + BB * DD;         // [64,512]

    float* xprojF = (float*)d_ws;                     // [512,96]
    float* xprojB = xprojF + KK * G3H;                // [512,96]

    // 1) PE input projection via f32 WMMA (2 dirs x 32 M-tiles).
    gpo_xproj_wmma<<<64, 32, 0, stream>>>(fwd_k, fwd_b, bwd_k, bwd_b,
                                          xprojF, xprojB);

    // 2) Bi-GRU recurrence + scores + masked softmax -> weights.
    gpo_gru_softmax<<<BB, G3H, 0, stream>>>(xprojF, xprojB,
                                            fwd_rk, fwd_b, bwd_rk, bwd_b,
                                            linear_w, lengths, wout);

    // 3) Per-(b,d) descending sort + weighted pooling.
    gpo_sort_pool<<<BB * DD, 256, 0, stream>>>(features, lengths, wout, pooled);
}